// CONV_COMPLEX_v1_5995774345826
// MI455X (gfx1250) — compile-verified
//
#include <hip/hip_runtime.h>
#include <math.h>

// ---------------------------------------------------------------------------
// Types
// ---------------------------------------------------------------------------
typedef __attribute__((ext_vector_type(2))) float v2f;
typedef __attribute__((ext_vector_type(8))) float v8f;

#define BATCH 4096
#define NCH   14
#define SLOPE 0.05f

// complex helpers -----------------------------------------------------------
__device__ __forceinline__ float2 cmac(float2 acc, float2 a, float2 b) {
    acc.x = fmaf(a.x, b.x, fmaf(-a.y, b.y, acc.x));
    acc.y = fmaf(a.x, b.y, fmaf(a.y, b.x, acc.y));
    return acc;
}
__device__ __forceinline__ float2 cleaky(float2 v) {
    float amp = sqrtf(v.x * v.x + v.y * v.y);      // |z| >= 0 always
    float ang = atan2f(v.y, v.x);
    ang = (ang < 0.0f) ? SLOPE * ang : ang;
    float s, c;
    __sincosf(ang, &s, &c);
    return make_float2(amp * c, amp * s);
}
__device__ __forceinline__ float2 cmaxc(float2 a, float2 b) {
    return make_float2(fmaxf(a.x, b.x), fmaxf(a.y, b.y));
}

// ---------------------------------------------------------------------------
// Repack: deinterleave complex weights into planar R / I planes (workspace).
// ---------------------------------------------------------------------------
__global__ void __launch_bounds__(256) repack_kernel(
    const float2* __restrict__ src, float* __restrict__ dR,
    float* __restrict__ dI, int count)
{
    const int i = blockIdx.x * blockDim.x + threadIdx.x;
    if (i < count) {
        const float2 v = src[i];
        dR[i] = v.x;
        dI[i] = v.y;
    }
}

// ---------------------------------------------------------------------------
// Kernel 1: group construction + conv32 + 3-layer circular conv stack.
// One wave32 per sequence (b, c); 4 waves per block; weights staged in LDS.
// Writes features PLANAR in X_sum layout: fR/fI[b*784 + f*14 + c].
// ---------------------------------------------------------------------------
__global__ void __launch_bounds__(128) feat_kernel(
    const float*  __restrict__ x,
    const float2* __restrict__ w32, const float2* __restrict__ b32,
    const float2* __restrict__ w1,  const float2* __restrict__ b1,
    const float2* __restrict__ w2,  const float2* __restrict__ b2,
    const float2* __restrict__ w3,  const float2* __restrict__ b3,
    float* __restrict__ fR, float* __restrict__ fI)
{
    __shared__ float2 s_w32[8 * 32];
    __shared__ float2 s_b32[8];
    __shared__ float2 s_w1[16 * 3];
    __shared__ float2 s_b1[16];
    __shared__ float2 s_w2[32 * 16 * 3];
    __shared__ float2 s_b2[32];
    __shared__ float2 s_w3[32 * 32 * 3];
    __shared__ float2 s_b3[32];
    __shared__ float2 s_z[4][64];       // per-wave complex signal
    __shared__ float2 s_h1[4][16 * 16]; // after layer1 + pool2
    __shared__ float2 s_h2[4][32 * 8];  // after layer2 + pool2

    const int tid = threadIdx.x;
    for (int i = tid; i < 8 * 32; i += 128)      s_w32[i] = w32[i];
    for (int i = tid; i < 16 * 3; i += 128)      s_w1[i]  = w1[i];
    for (int i = tid; i < 32 * 16 * 3; i += 128) s_w2[i]  = w2[i];
    for (int i = tid; i < 32 * 32 * 3; i += 128) s_w3[i]  = w3[i];
    if (tid < 8)  s_b32[tid] = b32[tid];
    if (tid < 16) s_b1[tid]  = b1[tid];
    if (tid < 32) s_b2[tid]  = b2[tid];
    if (tid < 32) s_b3[tid]  = b3[tid];

    const int wave = tid >> 5;
    const int lane = tid & 31;
    const int seq  = blockIdx.x * 4 + wave;
    const int b    = seq / NCH;
    const int c    = seq - b * NCH;

    // ---- build complex signal z[0..63] from raw channels -------------------
    for (int tt = 0; tt < 2; ++tt) {
        const int t = lane + 32 * tt;
        const float* xr = x + (size_t)b * (64 * NCH) + (size_t)t * NCH;
        float re, im;
        if (c < 3)       { re = xr[c];     im = xr[c + 3]; }
        else if (c == 3) { re = 0.0f;      im = xr[6]; }
        else if (c < 7)  { re = xr[c - 4]; im = xr[c + 3]; }
        else if (c == 7) { re = 0.0f;      im = xr[10]; }
        else {
            int j  = (c < 11) ? (c - 8) : (c - 11);
            int j2 = (j == 2) ? 0 : (j + 1);
            re = xr[j] - xr[j2];
            im = (c < 11) ? xr[13] : xr[11 + (c - 11)];
        }
        s_z[wave][t] = make_float2(re, im);
    }
    __syncthreads();

    // ---- conv32 (kernel 32, stride 16) -> 8 filters x 3 positions ----------
    if (lane < 24) {
        const int f = lane / 3, p = lane - 3 * f;
        float2 acc = s_b32[f];
        #pragma unroll 8
        for (int k = 0; k < 32; ++k)
            acc = cmac(acc, s_z[wave][p * 16 + k], s_w32[f * 32 + k]);
        acc = cleaky(acc);
        const size_t o = (size_t)b * 784 + (size_t)lane * NCH + c; // fi = f*3+p
        fR[o] = acc.x;
        fI[o] = acc.y;
    }
    __syncthreads();

    // ---- layer 1: 1->16 ch, len 32 circular, cleaky, pool2 (fused) ---------
    for (int idx = lane; idx < 16 * 16; idx += 32) {
        const int ch = idx >> 4, p = idx & 15;
        float2 best = make_float2(-INFINITY, -INFINITY);
        for (int qq = 0; qq < 2; ++qq) {
            const int q = 2 * p + qq;
            float2 acc = s_b1[ch];
            #pragma unroll
            for (int k = 0; k < 3; ++k)
                acc = cmac(acc, s_z[wave][32 + ((q + k + 31) & 31)], s_w1[ch * 3 + k]);
            best = cmaxc(best, cleaky(acc));
        }
        s_h1[wave][ch * 16 + p] = best;
    }
    __syncthreads();

    // ---- layer 2: 16->32 ch, len 16 circular, cleaky, pool2 (fused) --------
    for (int idx = lane; idx < 32 * 8; idx += 32) {
        const int ch = idx >> 3, p = idx & 7;
        float2 best = make_float2(-INFINITY, -INFINITY);
        for (int qq = 0; qq < 2; ++qq) {
            const int q = 2 * p + qq;
            float2 acc = s_b2[ch];
            for (int ic = 0; ic < 16; ++ic) {
                #pragma unroll
                for (int k = 0; k < 3; ++k)
                    acc = cmac(acc, s_h1[wave][ic * 16 + ((q + k + 15) & 15)],
                               s_w2[(ch * 16 + ic) * 3 + k]);
            }
            best = cmaxc(best, cleaky(acc));
        }
        s_h2[wave][ch * 8 + p] = best;
    }
    __syncthreads();

    // ---- layer 3: 32->32 ch, len 8 circular, cleaky, pool8 -----------------
    {
        const int ch = lane; // 32 lanes == 32 channels
        float2 best = make_float2(-INFINITY, -INFINITY);
        for (int q = 0; q < 8; ++q) {
            float2 acc = s_b3[ch];
            for (int ic = 0; ic < 32; ++ic) {
                #pragma unroll
                for (int k = 0; k < 3; ++k)
                    acc = cmac(acc, s_h2[wave][ic * 8 + ((q + k + 7) & 7)],
                               s_w3[(ch * 32 + ic) * 3 + k]);
            }
            best = cmaxc(best, cleaky(acc));
        }
        const size_t o = (size_t)b * 784 + (size_t)(24 + ch) * NCH + c;
        fR[o] = best.x;
        fI[o] = best.y;
    }
}

// ---------------------------------------------------------------------------
// WMMA f32 16x16x4 wrapper.
// ---------------------------------------------------------------------------
__device__ __forceinline__ v8f wmma4(v2f a, v2f b, v8f c) {
    return __builtin_amdgcn_wmma_f32_16x16x4_f32(
        false, a, false, b, (short)0, c, false, false);
}
__device__ __forceinline__ v2f ldp(const float* p) { return *(const v2f*)p; }

// ---------------------------------------------------------------------------
// Layer-1 specialized complex GEMM: K=784, N=40 (3 N-tiles per wave).
// One wave per (16-row M tile, head); A loaded once, 12 WMMAs per K-step.
// Planar operands -> every load is a b64 straight into a WMMA VGPR pair.
// ---------------------------------------------------------------------------
__global__ void __launch_bounds__(256) gemm1_kernel(
    const float* __restrict__ AR, const float* __restrict__ AI,
    const float* __restrict__ WR, const float* __restrict__ WI,
    const float2* __restrict__ bias,
    float* __restrict__ OR_, float* __restrict__ OI_)
{
    const int wave = threadIdx.x >> 5;
    const int lane = threadIdx.x & 31;
    const int gid  = blockIdx.x * 8 + wave;     // 768 waves total
    const int mt   = gid / 3;
    const int h    = gid - mt * 3;

    const int m0   = mt * 16;
    const int lr   = lane & 15;
    const int hi   = lane >> 4;
    const int koff = hi * 2;

    const float* ArR = AR + (long)(m0 + lr) * 784 + koff;
    const float* ArI = AI + (long)(m0 + lr) * 784 + koff;
    const float* BrR[3];
    const float* BrI[3];
    #pragma unroll
    for (int t = 0; t < 3; ++t) {
        const int n = t * 16 + lr;
        const int nc = (n < 40) ? n : 39;
        BrR[t] = WR + ((long)h * 40 + nc) * 784 + koff;
        BrI[t] = WI + ((long)h * 40 + nc) * 784 + koff;
    }

    v8f aRR[3] = {}, aRI[3] = {}, aIR[3] = {}, aII[3] = {};
    for (int k0 = 0; k0 < 784; k0 += 4) {
        const v2f ar = ldp(ArR + k0);
        const v2f ai = ldp(ArI + k0);
        #pragma unroll
        for (int t = 0; t < 3; ++t) {
            const v2f br = ldp(BrR[t] + k0);
            const v2f bi = ldp(BrI[t] + k0);
            aRR[t] = wmma4(ar, br, aRR[t]);
            aII[t] = wmma4(ai, bi, aII[t]);
            aRI[t] = wmma4(ar, bi, aRI[t]);
            aIR[t] = wmma4(ai, br, aIR[t]);
        }
    }

    #pragma unroll
    for (int t = 0; t < 3; ++t) {
        const int n = t * 16 + lr;
        if (n < 40) {
            const float2 bb = bias[h * 40 + n];
            #pragma unroll
            for (int r = 0; r < 8; ++r) {
                const int m = r + hi * 8;
                float2 z = make_float2(aRR[t][r] - aII[t][r] + bb.x,
                                       aRI[t][r] + aIR[t][r] + bb.y);
                z = cleaky(z);
                const long o = ((long)h * BATCH + m0 + m) * 40 + n;
                OR_[o] = z.x;
                OI_[o] = z.y;
            }
        }
    }
}

// ---------------------------------------------------------------------------
// Generic planar complex GEMM layer (layers 2-4):
// Out[h,b,n] = cleaky( sum_i In[h,b,i]*W[h,n,i] + bias[h,n] )
// One wave per (16-row M tile, head, 16-col N tile).
// ---------------------------------------------------------------------------
__global__ void __launch_bounds__(256) cgemm_kernel(
    const float* __restrict__ AR, const float* __restrict__ AI,
    const float* __restrict__ WR, const float* __restrict__ WI,
    const float2* __restrict__ bias,
    float* __restrict__ OR_, float* __restrict__ OI_,
    int K, int N, int n_tiles, long in_head_stride)
{
    const int wave = threadIdx.x >> 5;
    const int lane = threadIdx.x & 31;
    const int gid  = blockIdx.x * 8 + wave;
    const int per_m = 3 * n_tiles;
    const int mt  = gid / per_m;
    const int rem = gid - mt * per_m;
    const int h   = rem / n_tiles;
    const int nt  = rem - h * n_tiles;

    const int m0   = mt * 16;
    const int lr   = lane & 15;
    const int hi   = lane >> 4;
    const int koff = hi * 2;
    const int n    = nt * 16 + lr;
    const bool nv  = (n < N);

    const float* ArR = AR + (long)h * in_head_stride + (long)(m0 + lr) * K + koff;
    const float* ArI = AI + (long)h * in_head_stride + (long)(m0 + lr) * K + koff;
    const long  woff = ((long)h * N + (nv ? n : 0)) * (long)K + koff;
    const float* BrR = WR + woff;
    const float* BrI = WI + woff;

    v8f aRR = {}, aRI = {}, aIR = {}, aII = {};
    for (int k0 = 0; k0 < K; k0 += 4) {
        const v2f ar = ldp(ArR + k0);
        const v2f ai = ldp(ArI + k0);
        const v2f br = ldp(BrR + k0);
        const v2f bi = ldp(BrI + k0);
        aRR = wmma4(ar, br, aRR);
        aII = wmma4(ai, bi, aII);
        aRI = wmma4(ar, bi, aRI);
        aIR = wmma4(ai, br, aIR);
    }

    if (nv) {
        const float2 bb = bias[h * N + n];
        #pragma unroll
        for (int r = 0; r < 8; ++r) {
            const int m = r + hi * 8;
            float2 z = make_float2(aRR[r] - aII[r] + bb.x,
                                   aRI[r] + aIR[r] + bb.y);
            z = cleaky(z);
            const long o = ((long)h * BATCH + m0 + m) * N + n;
            OR_[o] = z.x;
            OI_[o] = z.y;
        }
    }
}

// ---------------------------------------------------------------------------
// Final 20-tap complex dot + sigmoid(real) -> out[b*3 + h]
// ---------------------------------------------------------------------------
__global__ void __launch_bounds__(256) head_kernel(
    const float* __restrict__ zR, const float* __restrict__ zI,
    const float2* __restrict__ w5, const float2* __restrict__ b5,
    float* __restrict__ out)
{
    const int i = blockIdx.x * blockDim.x + threadIdx.x;
    if (i >= BATCH * 3) return;
    const int b = i / 3;
    const int h = i - b * 3;
    float2 acc = b5[h];
    const long base = ((long)h * BATCH + b) * 20;
    const float2* wr = w5 + h * 20;
    #pragma unroll 4
    for (int k = 0; k < 20; ++k)
        acc = cmac(acc, make_float2(zR[base + k], zI[base + k]), wr[k]);
    out[b * 3 + h] = 1.0f / (1.0f + __expf(-acc.x));
}

// ---------------------------------------------------------------------------
// Launch
// ---------------------------------------------------------------------------
extern "C" void kernel_launch(void* const* d_in, const int* in_sizes, int n_in,
                              void* d_out, int out_size, void* d_ws, size_t ws_size,
                              hipStream_t stream) {
    const float*  x    = (const float*)d_in[0];
    const float2* w32  = (const float2*)d_in[1];
    const float2* b32  = (const float2*)d_in[2];
    const float2* c3w1 = (const float2*)d_in[3];
    const float2* c3b1 = (const float2*)d_in[4];
    const float2* c3w2 = (const float2*)d_in[5];
    const float2* c3b2 = (const float2*)d_in[6];
    const float2* c3w3 = (const float2*)d_in[7];
    const float2* c3b3 = (const float2*)d_in[8];
    const float2* hw1  = (const float2*)d_in[9];
    const float2* hb1  = (const float2*)d_in[10];
    const float2* hw2  = (const float2*)d_in[11];
    const float2* hb2  = (const float2*)d_in[12];
    const float2* hw3  = (const float2*)d_in[13];
    const float2* hb3  = (const float2*)d_in[14];
    const float2* hw4  = (const float2*)d_in[15];
    const float2* hb4  = (const float2*)d_in[16];
    const float2* hw5  = (const float2*)d_in[17];
    const float2* hb5  = (const float2*)d_in[18];

    // --- workspace layout (floats), all planar R/I planes -------------------
    float* ws = (float*)d_ws;
    size_t cur = 0;
    const size_t nFeat = (size_t)BATCH * 784;
    const size_t nZ1 = (size_t)3 * BATCH * 40;
    const size_t nZ2 = (size_t)3 * BATCH * 160;
    const size_t nZ3 = (size_t)3 * BATCH * 40;
    const size_t nZ4 = (size_t)3 * BATCH * 20;
    const int nW1 = 3 * 40 * 784, nW2 = 3 * 160 * 40, nW3 = 3 * 40 * 160, nW4 = 3 * 20 * 40;

    float* fR  = ws + cur; cur += nFeat;
    float* fI  = ws + cur; cur += nFeat;
    float* z1R = ws + cur; cur += nZ1;
    float* z1I = ws + cur; cur += nZ1;
    float* z2R = ws + cur; cur += nZ2;
    float* z2I = ws + cur; cur += nZ2;
    float* z3R = ws + cur; cur += nZ3;
    float* z3I = ws + cur; cur += nZ3;
    float* z4R = ws + cur; cur += nZ4;
    float* z4I = ws + cur; cur += nZ4;
    float* w1R = ws + cur; cur += nW1;
    float* w1I = ws + cur; cur += nW1;
    float* w2R = ws + cur; cur += nW2;
    float* w2I = ws + cur; cur += nW2;
    float* w3R = ws + cur; cur += nW3;
    float* w3I = ws + cur; cur += nW3;
    float* w4R = ws + cur; cur += nW4;
    float* w4I = ws + cur; cur += nW4;

    // --- deinterleave MLP weights into planar planes -------------------------
    repack_kernel<<<(nW1 + 255) / 256, 256, 0, stream>>>(hw1, w1R, w1I, nW1);
    repack_kernel<<<(nW2 + 255) / 256, 256, 0, stream>>>(hw2, w2R, w2I, nW2);
    repack_kernel<<<(nW3 + 255) / 256, 256, 0, stream>>>(hw3, w3R, w3I, nW3);
    repack_kernel<<<(nW4 + 255) / 256, 256, 0, stream>>>(hw4, w4R, w4I, nW4);

    // --- features: 57344 sequences, 4 waves per 128-thread block -------------
    feat_kernel<<<(BATCH * NCH) / 4, 128, 0, stream>>>(
        x, w32, b32, c3w1, c3b1, c3w2, c3b2, c3w3, c3b3, fR, fI);

    // --- layer1: K=784, N=40, all 3 N-tiles per wave. 768 waves -> 96 blocks -
    gemm1_kernel<<<96, 256, 0, stream>>>(fR, fI, w1R, w1I, hb1, z1R, z1I);
    // --- layer2: K=40,  N=160 (10 n-tiles). 7680 waves -> 960 blocks ---------
    cgemm_kernel<<<960, 256, 0, stream>>>(z1R, z1I, w2R, w2I, hb2, z2R, z2I,
                                          40, 160, 10, (long)BATCH * 40);
    // --- layer3: K=160, N=40 (3 n-tiles). 2304 waves -> 288 blocks -----------
    cgemm_kernel<<<288, 256, 0, stream>>>(z2R, z2I, w3R, w3I, hb3, z3R, z3I,
                                          160, 40, 3, (long)BATCH * 160);
    // --- layer4: K=40,  N=20 (2 n-tiles). 1536 waves -> 192 blocks -----------
    cgemm_kernel<<<192, 256, 0, stream>>>(z3R, z3I, w4R, w4I, hb4, z4R, z4I,
                                          40, 20, 2, (long)BATCH * 40);

    // --- final dot + sigmoid: 12288 threads ----------------------------------
    head_kernel<<<48, 256, 0, stream>>>(z4R, z4I, hw5, hb5, (float*)d_out);
}